// DGCNN_8839042695322
// MI455X (gfx1250) — compile-verified
//
#include <hip/hip_runtime.h>

typedef __attribute__((ext_vector_type(2))) float v2f;
typedef __attribute__((ext_vector_type(8))) float v8f;

// SH / Zernike constants
#define C0f   0.28209479177387814f
#define C1f   0.4886025119029199f
#define C2XYf 1.0925484305920792f
#define C20f  0.31539156525252005f
#define C22f  0.5462742152960396f
#define C33f  0.5900435899266435f
#define C32f  2.890611442640554f
#define C31f  0.4570457994644658f
#define C30f  0.3731763325901154f
#define C3P2f 1.445305721320277f

// M[k][j] entry of the folded 20x64 weight matrix, output column j = m*4+u.
// Bias folded into k==0 (constant-1 feature).
__device__ __forceinline__ float mval(int k, int m,
                                      float w00u, float w01u, float b0u,
                                      float w10u, float w11u, float w2u, float w3u) {
    if (k == 0)  return (m == 0)     ? fmaf(C0f, w00u, b0u) : 0.0f;
    if (k == 1)  return (m == 0)     ? (C0f * w01u)         : 0.0f;
    if (k <= 4)  return (m == k - 1) ? (C1f * w10u)         : 0.0f;  // py,pz,px -> m=1..3
    if (k <= 7)  return (m == k - 4) ? (C1f * w11u)         : 0.0f;  // n2*{py,pz,px}
    if (k <= 12) return (m == k - 4) ? w2u                  : 0.0f;  // Y2 -> m=4..8
    return              (m == k - 4) ? w3u                  : 0.0f;  // Y3 -> m=9..15
}

// The 5 non-trivial (tile, kbase) WMMA steps of the block-structured GEMM:
//   step:   0     1     2     3     4
//   tile:   0     0     1     2     3
//   kbase:  0     4     8     12    16
__constant__ const int kStepTile[5]  = {0, 0, 1, 2, 3};
__constant__ const int kStepKbase[5] = {0, 4, 8, 12, 16};

// Main kernel: processes only FULL 16-point groups — no guards anywhere.
__global__ __launch_bounds__(256) void zernike_wmma_kernel(
    const float* __restrict__ x,
    const float* __restrict__ W0, const float* __restrict__ b0,
    const float* __restrict__ W1, const float* __restrict__ W2,
    const float* __restrict__ W3,
    float* __restrict__ out,
    int num_full_groups, int wave_stride) {

    const int lane   = threadIdx.x & 31;
    const int lane15 = lane & 15;
    const int half   = lane >> 4;     // 0: lanes 0-15, 1: lanes 16-31
    const int u      = lane15 & 3;    // output unit for this lane's columns

    // Per-lane weight scalars for unit u
    const float w00u = W0[u], w01u = W0[4 + u], b0u = b0[u];
    const float w10u = W1[u], w11u = W1[4 + u];
    const float w2u  = W2[u], w3u  = W3[u];

    // B-matrix registers for the 5 steps:
    //   Bv[step][v] = M[kbase(step) + v + 2*half][16*tile(step) + lane15]
    // (B 4x16 f32 layout: K = vgpr + 2*half, N = lane&15)
    float Bv[5][2];
#pragma unroll
    for (int s = 0; s < 5; ++s) {
        const int m = 4 * kStepTile[s] + (lane15 >> 2);
#pragma unroll
        for (int v = 0; v < 2; ++v) {
            const int k = kStepKbase[s] + v + 2 * half;
            Bv[s][v] = mval(k, m, w00u, w01u, b0u, w10u, w11u, w2u, w3u);
        }
    }

    const int wave_id = blockIdx.x * (blockDim.x >> 5) + (threadIdx.x >> 5);

    for (int g = wave_id; g < num_full_groups; g += wave_stride) {
        const int p = g * 16 + lane15;

        const float* xp = x + (size_t)p * 3;
        const float px = xp[0];
        const float py = xp[1];
        const float pz = xp[2];

        // Prefetch next group's coordinates (global_prefetch_b8)
        if (g + wave_stride < num_full_groups)
            __builtin_prefetch(x + (size_t)(p + 16 * wave_stride) * 3, 0, 1);

        const float x2 = px * px, y2 = py * py, z2 = pz * pz;
        const float n2 = x2 + y2 + z2;

        // 20 features (A-matrix rows for this lane's point M = lane15)
        float f[20];
        f[0]  = 1.0f;
        f[1]  = n2;
        f[2]  = py;        f[3]  = pz;        f[4]  = px;
        f[5]  = n2 * py;   f[6]  = n2 * pz;   f[7]  = n2 * px;
        f[8]  = C2XYf * px * py;
        f[9]  = C2XYf * py * pz;
        f[10] = C20f * (2.0f * z2 - x2 - y2);
        f[11] = C2XYf * px * pz;
        f[12] = C22f * (x2 - y2);
        f[13] = C33f  * py * (3.0f * x2 - y2);
        f[14] = C32f  * px * py * pz;
        f[15] = C31f  * py * (4.0f * z2 - x2 - y2);
        f[16] = C30f  * pz * (2.0f * z2 - 3.0f * x2 - 3.0f * y2);
        f[17] = C31f  * px * (4.0f * z2 - x2 - y2);
        f[18] = C3P2f * pz * (x2 - y2);
        f[19] = C33f  * px * (x2 - 3.0f * y2);

        const int base_row = g * 16 + 8 * half;

        // A operands: A[v] = f[kbase + v + 2*half]  (A 16x4 f32 layout)
        v2f A[5];
#pragma unroll
        for (int s = 0; s < 5; ++s) {
            const int kb = kStepKbase[s];
            A[s][0] = half ? f[kb + 2] : f[kb + 0];
            A[s][1] = half ? f[kb + 3] : f[kb + 1];
        }

        v2f B0; B0[0] = Bv[0][0]; B0[1] = Bv[0][1];
        v2f B1; B1[0] = Bv[1][0]; B1[1] = Bv[1][1];
        v2f B2; B2[0] = Bv[2][0]; B2[1] = Bv[2][1];
        v2f B3; B3[0] = Bv[3][0]; B3[1] = Bv[3][1];
        v2f B4; B4[0] = Bv[4][0]; B4[1] = Bv[4][1];

        v8f zero = {};

        // Tile 0 (j=0..15): two accumulation steps (k=0..7)
        v8f d0 = __builtin_amdgcn_wmma_f32_16x16x4_f32(false, A[0], false, B0,
                                                       (short)0, zero, false, false);
        d0     = __builtin_amdgcn_wmma_f32_16x16x4_f32(false, A[1], false, B1,
                                                       (short)0, d0, false, false);
        // Tiles 1..3 (j=16..63): single step each
        v8f d1 = __builtin_amdgcn_wmma_f32_16x16x4_f32(false, A[2], false, B2,
                                                       (short)0, zero, false, false);
        v8f d2 = __builtin_amdgcn_wmma_f32_16x16x4_f32(false, A[3], false, B3,
                                                       (short)0, zero, false, false);
        v8f d3 = __builtin_amdgcn_wmma_f32_16x16x4_f32(false, A[4], false, B4,
                                                       (short)0, zero, false, false);

        // D layout: VGPR r holds row M = r + 8*half, col N = lane15.
        // out flat index = point*64 + (16*tile + N). Unguarded non-temporal
        // streaming stores (write-once output; keep it out of L2).
        float* op = out + (size_t)base_row * 64 + (size_t)lane15;
#pragma unroll
        for (int r = 0; r < 8; ++r) {
            __builtin_nontemporal_store(d0[r], op + (size_t)r * 64 + 0);
            __builtin_nontemporal_store(d1[r], op + (size_t)r * 64 + 16);
            __builtin_nontemporal_store(d2[r], op + (size_t)r * 64 + 32);
            __builtin_nontemporal_store(d3[r], op + (size_t)r * 64 + 48);
        }
    }
}

// Scalar tail kernel: handles the (never-taken in this problem) remainder
// points with one thread per output element.
__global__ void zernike_tail_kernel(
    const float* __restrict__ x,
    const float* __restrict__ W0, const float* __restrict__ b0,
    const float* __restrict__ W1, const float* __restrict__ W2,
    const float* __restrict__ W3,
    float* __restrict__ out,
    int first_point, int num_points) {

    const int elems = (num_points - first_point) * 64;
    for (int i = blockIdx.x * blockDim.x + threadIdx.x; i < elems;
         i += gridDim.x * blockDim.x) {
        const int p = first_point + (i >> 6);
        const int j = i & 63;
        const int m = j >> 2, u = j & 3;

        const float px = x[(size_t)p * 3 + 0];
        const float py = x[(size_t)p * 3 + 1];
        const float pz = x[(size_t)p * 3 + 2];
        const float x2 = px * px, y2 = py * py, z2 = pz * pz;
        const float n2 = x2 + y2 + z2;

        float v;
        if (m == 0) {
            v = C0f * (W0[u] + n2 * W0[4 + u]) + b0[u];
        } else if (m <= 3) {
            const float comp = (m == 1) ? py : ((m == 2) ? pz : px);
            v = C1f * comp * (W1[u] + n2 * W1[4 + u]);
        } else if (m <= 8) {
            float Y2v;
            switch (m - 4) {
                case 0: Y2v = C2XYf * px * py; break;
                case 1: Y2v = C2XYf * py * pz; break;
                case 2: Y2v = C20f * (2.0f * z2 - x2 - y2); break;
                case 3: Y2v = C2XYf * px * pz; break;
                default: Y2v = C22f * (x2 - y2); break;
            }
            v = Y2v * W2[u];
        } else {
            float Y3v;
            switch (m - 9) {
                case 0: Y3v = C33f * py * (3.0f * x2 - y2); break;
                case 1: Y3v = C32f * px * py * pz; break;
                case 2: Y3v = C31f * py * (4.0f * z2 - x2 - y2); break;
                case 3: Y3v = C30f * pz * (2.0f * z2 - 3.0f * x2 - 3.0f * y2); break;
                case 4: Y3v = C31f * px * (4.0f * z2 - x2 - y2); break;
                case 5: Y3v = C3P2f * pz * (x2 - y2); break;
                default: Y3v = C33f * px * (x2 - 3.0f * y2); break;
            }
            v = Y3v * W3[u];
        }
        out[(size_t)p * 64 + j] = v;
    }
}

extern "C" void kernel_launch(void* const* d_in, const int* in_sizes, int n_in,
                              void* d_out, int out_size, void* d_ws, size_t ws_size,
                              hipStream_t stream) {
    const float* x  = (const float*)d_in[0];
    const float* W0 = (const float*)d_in[1];
    const float* b0 = (const float*)d_in[2];
    const float* W1 = (const float*)d_in[3];
    const float* W2 = (const float*)d_in[4];
    const float* W3 = (const float*)d_in[5];
    float* out = (float*)d_out;

    const int num_points      = in_sizes[0] / 3;   // 1,048,576
    const int num_full_groups = num_points / 16;   // 65,536
    const int rem             = num_points - num_full_groups * 16;  // 0 here

    if (num_full_groups > 0) {
        const int threads = 256;                    // 8 waves
        const int waves_per_block = threads / 32;
        int blocks = (num_full_groups + waves_per_block - 1) / waves_per_block;
        if (blocks > 1024) blocks = 1024;           // grid-stride: ~8 groups/wave
        const int wave_stride = blocks * waves_per_block;

        zernike_wmma_kernel<<<blocks, threads, 0, stream>>>(
            x, W0, b0, W1, W2, W3, out, num_full_groups, wave_stride);
    }
    if (rem > 0) {
        zernike_tail_kernel<<<4, 256, 0, stream>>>(
            x, W0, b0, W1, W2, W3, out, num_full_groups * 16, num_points);
    }
}